// GCN_9998683865528
// MI455X (gfx1250) — compile-verified
//
#include <hip/hip_runtime.h>
#include <math.h>

typedef __attribute__((ext_vector_type(2))) float v2f;
typedef __attribute__((ext_vector_type(8))) float v8f;
typedef __attribute__((ext_vector_type(4))) unsigned int u32x4;
typedef __attribute__((ext_vector_type(8))) unsigned int u32x8;

#define N_NODES 50000
#define N_EDGES 800000
#define F_INDIM 128
#define HIDDIM  256
#define NCLS    64

// ---------------- degree ----------------
__global__ void k_init_deg(float* __restrict__ deg) {
  int i = blockIdx.x * 256 + threadIdx.x;
  if (i < N_NODES) deg[i] = 1.0f;  // self-loop contribution
}

__global__ void k_count_deg(const long long* __restrict__ dst, float* __restrict__ deg) {
  int e = blockIdx.x * 256 + threadIdx.x;
  if (e < N_EDGES) atomicAdd(&deg[(int)dst[e]], 1.0f);
}

__global__ void k_rsqrt_deg(float* __restrict__ deg) {
  int i = blockIdx.x * 256 + threadIdx.x;
  if (i < N_NODES) deg[i] = rsqrtf(deg[i]);  // deg >= 1 always (self-loops)
}

// ------------- self-loop scaled init: out[i,f] = in[i,f] * dis[i]^2 -------------
template <int F>
__global__ void k_self_scale(const float* __restrict__ in, const float* __restrict__ dis,
                             float* __restrict__ out) {
  int idx = blockIdx.x * 256 + threadIdx.x;
  if (idx >= N_NODES * F) return;
  int i = idx / F;
  float d = dis[i];
  out[idx] = in[idx] * d * d;
}

// ------------- edge aggregation: out[dst,f] += feat[src,f] * dis[s]*dis[d] -------------
template <int F, int EPB>
__global__ void k_edge_agg(const float* __restrict__ feat,
                           const long long* __restrict__ src,
                           const long long* __restrict__ dst,
                           const float* __restrict__ dis,
                           float* __restrict__ out) {
  const int slots = 256 / F;           // concurrent edges per block
  int slot = threadIdx.x / F;
  int f    = threadIdx.x & (F - 1);
  int e    = blockIdx.x * (slots * EPB) + slot;
#pragma unroll 4
  for (int it = 0; it < EPB; ++it, e += slots) {
    if (e >= N_EDGES) break;
    if (e + 8 * slots < N_EDGES) {     // pull edge list ahead -> global_prefetch_b8
      __builtin_prefetch(&src[e + 8 * slots], 0, 0);
      __builtin_prefetch(&dst[e + 8 * slots], 0, 0);
    }
    int s = (int)src[e];
    int d = (int)dst[e];
    float w = dis[s] * dis[d];
    atomicAdd(&out[d * F + f], feat[s * F + f] * w);
  }
}

// ------------- Tensor Data Mover: DMA a 2-D f32 tile (global -> LDS) -------------
// Builds the D# per CDNA5 ISA and issues TENSOR_LOAD_TO_LDS
// (2-group form: VADDR2/3 = NULL, tile up to 2-D). workgroup_mask = 0 (not in a cluster).
__device__ __forceinline__ void tdm_load_tile_f32(unsigned lds_byte_off,
                                                  const float* gptr,
                                                  unsigned tensor_d0, unsigned tensor_d1,
                                                  unsigned tile_d0,   unsigned tile_d1,
                                                  unsigned row_stride_elems) {
  unsigned long long ga = (unsigned long long)(uintptr_t)gptr;
  u32x4 g0;
  g0[0] = 1u;                                                 // count=1, user descriptor
  g0[1] = lds_byte_off;                                       // lds_addr (bytes)
  g0[2] = (unsigned)ga;                                       // global_addr[31:0]
  g0[3] = ((unsigned)(ga >> 32) & 0x01FFFFFFu) | (2u << 30);  // global_addr[56:32] | type=2
  u32x8 g1;
  g1[0] = 2u << 16;                                           // wg_mask=0, data_size=2 (4 B)
  g1[1] = (tensor_d0 & 0xFFFFu) << 16;                        // tensor_dim0[15:0]
  g1[2] = (tensor_d0 >> 16) | ((tensor_d1 & 0xFFFFu) << 16);  // td0[31:16] | td1[15:0]
  g1[3] = (tensor_d1 >> 16) | ((tile_d0 & 0xFFFFu) << 16);    // td1[31:16] | tile_dim0
  g1[4] = tile_d1 & 0xFFFFu;                                  // tile_dim1 | tile_dim2=0
  g1[5] = row_stride_elems;                                   // tensor_dim0_stride[31:0]
  g1[6] = 0u;                                                 // stride0[47:32] | stride1 lo
  g1[7] = 0u;                                                 // stride1 hi
  asm volatile("tensor_load_to_lds %0, %1" :: "s"(g0), "s"(g1) : "memory");
}

// ------------- WMMA f32 GEMM: C[n x NCOL] = A[n x K] @ B[K x NCOL] (+epilogue) -------------
// B panel (K x PANEL) staged into LDS by the TDM; one wave computes a 16x16 tile via
// V_WMMA_F32_16X16X4_F32. C/D layout (ISA 7.12.2): VGPR r -> row (r + 8*laneHalf), col=lane&15.
template <int K, int NCOL, int PANEL, int WAVES, bool FUSE_EPI>
__global__ void k_gemm_wmma(const float* __restrict__ A,
                            const float* __restrict__ B,
                            const float* __restrict__ bias,
                            const float* __restrict__ dmask,
                            float* __restrict__ C) {
  __shared__ float Bs[K * PANEL];      // 64 KB panel (fits in 320 KB WGP LDS)
  int lane = threadIdx.x & 31;
  int wave = threadIdx.x >> 5;
  int half = lane >> 4;                // 0: lanes 0-15, 1: lanes 16-31
  int sub  = lane & 15;
  int rowBase   = blockIdx.x * 16;
  int panelBase = blockIdx.y * PANEL;

  if (wave == 0) {
    tdm_load_tile_f32((unsigned)(uintptr_t)(void*)Bs, B + panelBase,
                      /*tensor_d0=*/NCOL, /*tensor_d1=*/K,
                      /*tile_d0=*/PANEL,  /*tile_d1=*/K,
                      /*stride0=*/NCOL);
    __builtin_amdgcn_s_wait_tensorcnt(0);
  }
  __syncthreads();

  const float* Arow = A + (rowBase + sub) * K;
  const float* Bw   = Bs + wave * 16 + sub;
  v8f c = {};
#pragma unroll 8
  for (int k = 0; k < K; k += 4) {
    int kb = k + 2 * half;             // lanes 0-15: K={k,k+1}; lanes 16-31: K={k+2,k+3}
    v2f a, b;
    a.x = Arow[kb];
    a.y = Arow[kb + 1];
    b.x = Bw[kb * PANEL];
    b.y = Bw[(kb + 1) * PANEL];
    c = __builtin_amdgcn_wmma_f32_16x16x4_f32(
        /*neg_a=*/false, a, /*neg_b=*/false, b,
        /*c_mod=*/(short)0, c, /*reuse_a=*/false, /*reuse_b=*/false);
  }
#pragma unroll
  for (int r = 0; r < 8; ++r) {
    int row = rowBase + r + 8 * half;
    int col = panelBase + wave * 16 + sub;
    float v = c[r];
    if constexpr (FUSE_EPI) {
      v += bias[col];
      v = fmaxf(v, 0.0f);
      v *= dmask[row * NCOL + col];
    }
    C[row * NCOL + col] = v;
  }
}

// ------------- epilogue: out = log_softmax(out + b2), one wave32 per row -------------
__global__ void k_logsoftmax(float* __restrict__ out, const float* __restrict__ b2) {
  int lane = threadIdx.x & 31;
  int warp = threadIdx.x >> 5;
  int row  = blockIdx.x * 8 + warp;
  if (row >= N_NODES) return;
  float v0 = out[row * NCLS + lane]      + b2[lane];
  float v1 = out[row * NCLS + lane + 32] + b2[lane + 32];
  float m = fmaxf(v0, v1);
#pragma unroll
  for (int off = 16; off; off >>= 1) m = fmaxf(m, __shfl_xor(m, off, 32));
  float s = expf(v0 - m) + expf(v1 - m);
#pragma unroll
  for (int off = 16; off; off >>= 1) s += __shfl_xor(s, off, 32);
  float l = logf(s);
  out[row * NCLS + lane]      = v0 - m - l;
  out[row * NCLS + lane + 32] = v1 - m - l;
}

extern "C" void kernel_launch(void* const* d_in, const int* in_sizes, int n_in,
                              void* d_out, int out_size, void* d_ws, size_t ws_size,
                              hipStream_t stream) {
  const float*      x     = (const float*)d_in[0];
  const long long*  ei    = (const long long*)d_in[1];
  const float*      W1    = (const float*)d_in[2];
  const float*      b1    = (const float*)d_in[3];
  const float*      W2    = (const float*)d_in[4];
  const float*      b2    = (const float*)d_in[5];
  const float*      dmask = (const float*)d_in[6];
  float*            out   = (float*)d_out;
  const long long*  src   = ei;
  const long long*  dst   = ei + N_EDGES;

  // workspace layout (floats): deg[50176] | xa[N*128] (reused as h2[N*64]) | h[N*256]
  float* deg = (float*)d_ws;
  float* xa  = deg + 50176;
  float* h   = xa + (size_t)N_NODES * F_INDIM;
  float* h2  = xa;  // N*64 fits in the N*128 slot once xa has been consumed by GEMM1

  // 1) symmetric normalization coefficients
  k_init_deg <<<(N_NODES + 255) / 256, 256, 0, stream>>>(deg);
  k_count_deg<<<(N_EDGES + 255) / 256, 256, 0, stream>>>(dst, deg);
  k_rsqrt_deg<<<(N_NODES + 255) / 256, 256, 0, stream>>>(deg);

  // 2) layer-1 aggregation in input space (128 feats): xa = self + edge scatter
  k_self_scale<F_INDIM><<<(N_NODES * F_INDIM + 255) / 256, 256, 0, stream>>>(x, deg, xa);
  k_edge_agg<F_INDIM, 16><<<(N_EDGES + 31) / 32, 256, 0, stream>>>(x, src, dst, deg, xa);

  // 3) h = dropout(relu(xa @ W1 + b1))   [TDM-staged B, WMMA f32, fused epilogue]
  {
    dim3 grid(N_NODES / 16, HIDDIM / 128);
    k_gemm_wmma<F_INDIM, HIDDIM, 128, 8, true><<<grid, 256, 0, stream>>>(xa, W1, b1, dmask, h);
  }

  // 4) h2 = h @ W2   [TDM-staged B, WMMA f32, bias deferred past aggregation]
  {
    dim3 grid(N_NODES / 16, 1);
    k_gemm_wmma<HIDDIM, NCLS, 64, 4, false><<<grid, 128, 0, stream>>>(h, W2, nullptr, nullptr, h2);
  }

  // 5) layer-2 aggregation in class space (64 feats) directly into d_out
  k_self_scale<NCLS><<<(N_NODES * NCLS + 255) / 256, 256, 0, stream>>>(h2, deg, out);
  k_edge_agg<NCLS, 16><<<(N_EDGES + 63) / 64, 256, 0, stream>>>(h2, src, dst, deg, out);

  // 6) out = log_softmax(out + b2)
  k_logsoftmax<<<N_NODES / 8, 256, 0, stream>>>(out, b2);
}